// RankerXQuad_73005854097667
// MI455X (gfx1250) — compile-verified
//
#include <hip/hip_runtime.h>
#include <stdint.h>

#define U_USERS   2048
#define N_ITEMS   25000
#define NV4       (N_ITEMS / 4)   // 6250
#define K_SEL     10
#define RMAX      100
#define HEADN     280
#define LMBDA     0.4f
#define NEGV      (-10000.0f)

typedef float v2f __attribute__((ext_vector_type(2)));
typedef float v8f __attribute__((ext_vector_type(8)));

#if defined(__gfx1250__) && __has_builtin(__builtin_amdgcn_wmma_f32_16x16x4_f32)
#define HAVE_WMMA_F32X4 1
#endif

// ---------------------------------------------------------------------------
// Kernel 1: head mask. Single block. Exact top-280 by (popularity desc, index
// asc), matching jnp.argsort(-pop) stability. Writes head_f32[N_ITEMS] to ws.
// ---------------------------------------------------------------------------
__device__ __forceinline__ int pop_bin(float v) {
    int b = (int)(v * 2048.0f);
    b = b < 0 ? 0 : (b > 2047 ? 2047 : b);
    return b;
}

__global__ __launch_bounds__(1024) void headmask_kernel(const float* __restrict__ pop,
                                                        float* __restrict__ head) {
    __shared__ float sp[N_ITEMS];
    __shared__ int   hist[2048];
    __shared__ int   eqIdx[1024];
    __shared__ int   chosen[HEADN];
    __shared__ int   s_tb, s_need, s_eqCount;
    const int tid = threadIdx.x;

    for (int i = tid; i < N_ITEMS; i += 1024) sp[i] = pop[i];
    for (int i = tid; i < 2048; i += 1024) hist[i] = 0;
    if (tid == 0) s_eqCount = 0;
    __syncthreads();

    for (int i = tid; i < N_ITEMS; i += 1024) atomicAdd(&hist[pop_bin(sp[i])], 1);
    __syncthreads();

    if (tid == 0) {
        int cum = 0, tb = 0;
        for (int b = 2047; b >= 0; --b) {
            if (cum + hist[b] >= HEADN) { tb = b; break; }
            cum += hist[b];
        }
        s_tb = tb;
        s_need = HEADN - cum;
    }
    __syncthreads();
    const int tb = s_tb;

    for (int i = tid; i < N_ITEMS; i += 1024) {
        if (pop_bin(sp[i]) == tb) {
            int p = atomicAdd(&s_eqCount, 1);
            if (p < 1024) eqIdx[p] = i;
        }
    }
    __syncthreads();

    if (tid == 0) {
        int cnt = s_eqCount; if (cnt > 1024) cnt = 1024;
        int need = s_need;   if (need > cnt) need = cnt;
        for (int t = 0; t < need; ++t) {
            int best = 0; float bv = -1e30f; int bi = 0x7fffffff;
            for (int j = 0; j < cnt; ++j) {
                int idx = eqIdx[j];
                if (idx < 0) continue;
                float v = sp[idx];
                if (v > bv || (v == bv && idx < bi)) { bv = v; bi = idx; best = j; }
            }
            chosen[t] = eqIdx[best];
            eqIdx[best] = -1;
        }
        s_need = need;
    }
    __syncthreads();
    const int need = s_need;

    for (int i = tid; i < N_ITEMS; i += 1024) {
        int b = pop_bin(sp[i]);
        float h = 0.0f;
        if (b > tb) h = 1.0f;
        else if (b == tb) {
            for (int t = 0; t < need; ++t)
                if (chosen[t] == i) { h = 1.0f; break; }
        }
        head[i] = h;
    }
}

// ---------------------------------------------------------------------------
// Kernel 2: per-user [obs_sum, obs.head] via WMMA GEMV.
// D(16x16) = A(16 users x 4 items) * B(4 x 16 [ones, head, 0...]).
// 16 users per block, 8 waves split K=25000, branch-free B build,
// 2-way unrolled with independent accumulators, cross-wave reduce in LDS.
// ---------------------------------------------------------------------------
__global__ __launch_bounds__(256) void stats_kernel(const float* __restrict__ obs,
                                                    const float* __restrict__ head,
                                                    float* __restrict__ stats) {
#ifdef HAVE_WMMA_F32X4
    __shared__ float red[8][32];
    const int u0   = blockIdx.x * 16;
    const int tid  = threadIdx.x;
    const int wave = tid >> 5;
    const int lane = tid & 31;
    const int row  = lane & 15;        // M (user within tile) for A
    const int n    = lane & 15;        // N (output column) for B/D
    const int koff = (lane >> 4) * 2;  // K sub-pair {0,1} or {2,3}
    const bool isOnes = (n == 0);
    const bool isHead = (n == 1);

    const float* arow = obs + (size_t)(u0 + row) * N_ITEMS;
    v8f c0 = {0.f, 0.f, 0.f, 0.f, 0.f, 0.f, 0.f, 0.f};
    v8f c1 = {0.f, 0.f, 0.f, 0.f, 0.f, 0.f, 0.f, 0.f};

    int s = wave;
    for (; s + 8 < NV4; s += 16) {
        const int ba0 = 4 * s + koff;
        const int ba1 = 4 * (s + 8) + koff;
        float2 av0 = *(const float2*)(arow + ba0);
        float2 av1 = *(const float2*)(arow + ba1);
        float2 hv0 = *(const float2*)(head + ba0);   // half-wave-uniform address
        float2 hv1 = *(const float2*)(head + ba1);
        v2f a0; a0[0] = av0.x; a0[1] = av0.y;
        v2f a1; a1[0] = av1.x; a1[1] = av1.y;
        v2f b0, b1;                                   // select, no branches
        b0[0] = isOnes ? 1.0f : (isHead ? hv0.x : 0.0f);
        b0[1] = isOnes ? 1.0f : (isHead ? hv0.y : 0.0f);
        b1[0] = isOnes ? 1.0f : (isHead ? hv1.x : 0.0f);
        b1[1] = isOnes ? 1.0f : (isHead ? hv1.y : 0.0f);
        c0 = __builtin_amdgcn_wmma_f32_16x16x4_f32(false, a0, false, b0, (short)0, c0, false, false);
        c1 = __builtin_amdgcn_wmma_f32_16x16x4_f32(false, a1, false, b1, (short)0, c1, false, false);
    }
    if (s < NV4) {
        const int ba = 4 * s + koff;
        float2 av = *(const float2*)(arow + ba);
        float2 hv = *(const float2*)(head + ba);
        v2f a; a[0] = av.x; a[1] = av.y;
        v2f b;
        b[0] = isOnes ? 1.0f : (isHead ? hv.x : 0.0f);
        b[1] = isOnes ? 1.0f : (isHead ? hv.y : 0.0f);
        c0 = __builtin_amdgcn_wmma_f32_16x16x4_f32(false, a, false, b, (short)0, c0, false, false);
    }
    c0 += c1;   // exact: integer-valued f32 partial sums

    // D layout: lanes 0-15 hold M=0..7 (vgpr m), lanes 16-31 hold M=8..15; N = lane&15.
    if (n < 2) {
        const int mbase = (lane < 16) ? 0 : 8;
        for (int m = 0; m < 8; ++m) red[wave][n * 16 + (mbase + m)] = c0[m];
    }
    __syncthreads();
    if (tid < 32) {
        float acc = 0.0f;
        for (int w = 0; w < 8; ++w) acc += red[w][tid];
        const int col = tid >> 4, m = tid & 15;
        stats[2 * (u0 + m) + col] = acc;
    }
#else
    // Fallback (also what the host pass parses): 16 users/block, 16 threads/user.
    __shared__ float r0[256], r1[256];
    const int u0 = blockIdx.x * 16;
    const int tid = threadIdx.x;
    const int ur = tid >> 4, ln = tid & 15;
    const float* arow = obs + (size_t)(u0 + ur) * N_ITEMS;
    float s0 = 0.f, s1 = 0.f;
    for (int i = ln; i < N_ITEMS; i += 16) { float o = arow[i]; s0 += o; s1 += o * head[i]; }
    r0[tid] = s0; r1[tid] = s1;
    __syncthreads();
    for (int off = 8; off > 0; off >>= 1) {
        if (ln < off) { r0[tid] += r0[tid + off]; r1[tid] += r1[tid + off]; }
        __syncthreads();
    }
    if (ln == 0) { stats[2 * (u0 + ur)] = r0[tid]; stats[2 * (u0 + ur) + 1] = r1[tid]; }
#endif
}

// ---------------------------------------------------------------------------
// Kernel 3: per-user rerank. One block per user, 256 threads, row in LDS.
// ---------------------------------------------------------------------------
struct SMem3 {
    float s[N_ITEMS];                 // score row (raw -> normalized/masked)
    int   hist[1024];
    float rA[256]; int rI[256]; float rB[256];
    int   candIdx[RMAX]; float candVal[RMAX]; float candHead[RMAX]; int selFlag[RMAX];
    int   eqIdx[256]; float eqVal[256];
    int   selItems[K_SEL]; float selVals[K_SEL];
    float sMin, sInv, pHead, pTail, cHead, nSel;
    int   candCount, eqCount, tb, needSel;
};

__global__ __launch_bounds__(256) void rerank_kernel(const float* __restrict__ pred,
                                                     const float* __restrict__ obs,
                                                     const float* __restrict__ head,
                                                     const float* __restrict__ stats,
                                                     float* __restrict__ out) {
    __shared__ SMem3 sm;
    const int tid = threadIdx.x;
    const int u   = blockIdx.x;
    const float* srow = pred + (size_t)u * N_ITEMS;
    const float* orow = obs  + (size_t)u * N_ITEMS;
    float*       wrow = out  + (size_t)u * N_ITEMS;

    // --- 1. stage score row into LDS (CDNA5 async-to-LDS path) ---
#if defined(__gfx1250__)
    for (int i = tid; i < NV4; i += 256) {
        unsigned lds_off = (unsigned)(uintptr_t)(&sm.s[4 * i]);
        const float* g = srow + 4 * i;
        asm volatile("global_load_async_to_lds_b128 %0, %1, off"
                     :: "v"(lds_off), "v"(g) : "memory");
    }
    asm volatile("s_wait_asynccnt 0" ::: "memory");
#else
    for (int i = tid; i < NV4; i += 256) {
        float4 v = *(const float4*)(srow + 4 * i);
        *(float4*)(&sm.s[4 * i]) = v;
    }
#endif
    __syncthreads();

    // --- 2. min / max reduction over raw scores ---
    float mn = 3.4e38f, mx = -3.4e38f;
    for (int i = tid; i < N_ITEMS; i += 256) {
        float v = sm.s[i];
        mn = fminf(mn, v);
        mx = fmaxf(mx, v);
    }
    sm.rA[tid] = mn; sm.rB[tid] = mx;
    __syncthreads();
    for (int off = 128; off > 0; off >>= 1) {
        if (tid < off) {
            sm.rA[tid] = fminf(sm.rA[tid], sm.rA[tid + off]);
            sm.rB[tid] = fmaxf(sm.rB[tid], sm.rB[tid + off]);
        }
        __syncthreads();
    }
    if (tid == 0) {
        sm.sMin = sm.rA[0];
        sm.sInv = 1.0f / (sm.rB[0] - sm.rA[0]);
        sm.candCount = 0; sm.eqCount = 0;
        sm.cHead = 0.0f;  sm.nSel = 0.0f;
        float osum = stats[2 * u], hsum = stats[2 * u + 1];
        float ph = hsum / osum;
        sm.pHead = ph; sm.pTail = 1.0f - ph;
    }
    for (int i = tid; i < 1024; i += 256) sm.hist[i] = 0;
    __syncthreads();
    const float smin = sm.sMin, sinv = sm.sInv;

    // --- 3. normalize, mask observed to NEG, histogram non-negative values ---
    for (int i = tid; i < NV4; i += 256) {
        float4 o = *(const float4*)(orow + 4 * i);
        float oo[4] = {o.x, o.y, o.z, o.w};
#pragma unroll
        for (int j = 0; j < 4; ++j) {
            float v = (sm.s[4 * i + j] - smin) * sinv;
            if (oo[j] > 0.0f) v = NEGV;
            sm.s[4 * i + j] = v;
            if (v >= 0.0f) {
                int b = (int)(v * 1024.0f);
                b = b > 1023 ? 1023 : b;
                atomicAdd(&sm.hist[b], 1);
            }
        }
    }
    __syncthreads();

    // --- 4. top-RMAX threshold bin ---
    if (tid == 0) {
        int cum = 0, tb = 1023;
        for (int b = 1023; b >= 0; --b) {
            if (cum + sm.hist[b] >= RMAX) { tb = b; break; }
            cum += sm.hist[b];
        }
        sm.tb = tb;
        sm.needSel = RMAX - cum;
    }
    __syncthreads();
    const int tb = sm.tb;

    // --- 5. collect candidates (strictly above bin + exact within-bin select) ---
    for (int i = tid; i < N_ITEMS; i += 256) {
        float v = sm.s[i];
        if (v < 0.0f) continue;
        int b = (int)(v * 1024.0f);
        b = b > 1023 ? 1023 : b;
        if (b > tb) {
            int p = atomicAdd(&sm.candCount, 1);
            sm.candIdx[p] = i; sm.candVal[p] = v;
        } else if (b == tb) {
            int p = atomicAdd(&sm.eqCount, 1);
            if (p < 256) { sm.eqIdx[p] = i; sm.eqVal[p] = v; }
        }
    }
    __syncthreads();
    if (tid == 0) {
        int cnt = sm.eqCount; if (cnt > 256) cnt = 256;
        int need = sm.needSel; if (need > cnt) need = cnt;
        int base = sm.candCount;
        for (int t = 0; t < need; ++t) {
            int best = 0; float bv = -1e30f; int bi = 0x7fffffff;
            for (int j = 0; j < cnt; ++j) {
                int idx = sm.eqIdx[j];
                if (idx < 0) continue;
                float v = sm.eqVal[j];
                if (v > bv || (v == bv && idx < bi)) { bv = v; bi = idx; best = j; }
            }
            sm.candIdx[base + t] = sm.eqIdx[best];
            sm.candVal[base + t] = bv;
            sm.eqIdx[best] = -1;
        }
        sm.candCount = base + need;
    }
    __syncthreads();
    const int ncand = sm.candCount;
    for (int t = tid; t < ncand; t += 256) {
        sm.candHead[t] = head[sm.candIdx[t]];
        sm.selFlag[t]  = 0;
    }
    __syncthreads();

    // --- 6. greedy xQuAD selection (K_SEL steps, block-wide argmax) ---
    const float ph = sm.pHead, pt = sm.pTail;
    for (int t = 0; t < K_SEL; ++t) {
        const float nSel  = sm.nSel;
        const float cH    = sm.cHead;
        const float denom = fmaxf(nSel, 1.0f);
        const float fh = (nSel > 0.0f) ? (1.0f - cH / denom) : 1.0f;
        const float ft = (nSel > 0.0f) ? (1.0f - (nSel - cH) / denom) : 1.0f;
        float comb = -1e30f;
        if (tid < ncand && !sm.selFlag[tid]) {
            float chd = sm.candHead[tid];
            float w = ph * chd * fh + pt * (1.0f - chd) * ft;
            comb = (1.0f - LMBDA) * sm.candVal[tid] + LMBDA * w;
        }
        sm.rA[tid] = comb; sm.rI[tid] = tid;
        __syncthreads();
        for (int off = 128; off > 0; off >>= 1) {
            if (tid < off) {
                float v2 = sm.rA[tid + off]; int i2 = sm.rI[tid + off];
                if (v2 > sm.rA[tid] || (v2 == sm.rA[tid] && i2 < sm.rI[tid])) {
                    sm.rA[tid] = v2; sm.rI[tid] = i2;
                }
            }
            __syncthreads();
        }
        if (tid == 0) {
            int best = sm.rI[0];
            sm.selFlag[best] = 1;
            sm.cHead = cH + sm.candHead[best];
            sm.nSel  = nSel + 1.0f;
            sm.selItems[t] = sm.candIdx[best];
            sm.selVals[t]  = (float)(K_SEL - (t + 1)) / (float)K_SEL;
        }
        __syncthreads();
    }

    // --- 7. write output row (zeros except the K_SEL selected positions) ---
    for (int i = tid; i < NV4; i += 256) {
        const int base = 4 * i;
        float o4[4] = {0.f, 0.f, 0.f, 0.f};
#pragma unroll
        for (int j = 0; j < K_SEL; ++j) {
            int d = sm.selItems[j] - base;
            if ((unsigned)d < 4u) o4[d] = sm.selVals[j];
        }
        float4 w4; w4.x = o4[0]; w4.y = o4[1]; w4.z = o4[2]; w4.w = o4[3];
        *(float4*)(wrow + base) = w4;
    }
}

// ---------------------------------------------------------------------------
extern "C" void kernel_launch(void* const* d_in, const int* in_sizes, int n_in,
                              void* d_out, int out_size, void* d_ws, size_t ws_size,
                              hipStream_t stream) {
    const float* pred = (const float*)d_in[0];
    const float* obs  = (const float*)d_in[1];
    const float* pop  = (const float*)d_in[2];
    float* out = (float*)d_out;

    float* head  = (float*)d_ws;            // N_ITEMS floats
    float* stats = (float*)d_ws + N_ITEMS;  // 2*U_USERS floats

    headmask_kernel<<<dim3(1), dim3(1024), 0, stream>>>(pop, head);
    stats_kernel<<<dim3(U_USERS / 16), dim3(256), 0, stream>>>(obs, head, stats);
    rerank_kernel<<<dim3(U_USERS), dim3(256), 0, stream>>>(pred, obs, head, stats, out);
}